// AGPFuser_90048284328257
// MI455X (gfx1250) — compile-verified
//
#include <hip/hip_runtime.h>
#include <math.h>

typedef __attribute__((ext_vector_type(2))) float v2f;
typedef __attribute__((ext_vector_type(8))) float v8f;

#if defined(__gfx1250__) && __has_builtin(__builtin_amdgcn_wmma_f32_16x16x4_f32)
#define USE_WMMA_F32 1
__device__ inline v8f wmma4(v2f a, v2f b, v8f c) {
  // V_WMMA_F32_16X16X4_F32 : D(16x16,f32) = A(16x4,f32) * B(4x16,f32) + C
  return __builtin_amdgcn_wmma_f32_16x16x4_f32(false, a, false, b, (short)0, c, false, false);
}
#endif

// ---------------- Tensor Data Mover (TDM) column tile load/store -------------
#if defined(__gfx1250__) && __has_builtin(__builtin_amdgcn_tensor_load_to_lds) && \
    __has_builtin(__builtin_amdgcn_tensor_store_from_lds) && \
    __has_builtin(__builtin_amdgcn_s_wait_tensorcnt)
#define HAVE_TDM 1
typedef __attribute__((ext_vector_type(4))) unsigned int v4u;
typedef __attribute__((ext_vector_type(4))) int v4i;
typedef __attribute__((ext_vector_type(8))) int v8i;

// One 1x512 tile of 8-byte elements (float2 column), row stride 512 elements.
__device__ __forceinline__ void tdm_col(unsigned lds_addr, unsigned long long ga, bool store) {
  v4u g0 = { 1u,                                            // count=1, user mode
             lds_addr,                                      // D#.lds_addr
             (unsigned)(ga & 0xffffffffull),                // global_addr[31:0]
             (unsigned)((ga >> 32) & 0x01ffffffull) | (2u << 30) };  // addr[56:32] | type=2
  v8i g1 = { (int)(3u << 16),      // workgroup_mask=0, data_size=3 (8 bytes)
             (int)(512u << 16),    // atomic_barrier_addr=0 | tensor_dim0=512 (lo16)
             (int)(512u << 16),    // tensor_dim0 hi=0 | tensor_dim1=512 (lo16)
             (int)(1u << 16),      // tensor_dim1 hi=0 | tile_dim0=1
             512,                  // tile_dim1=512 | tile_dim2=0
             512,                  // tensor_dim0_stride[31:0] = 512 elements
             0,                    // stride0 hi | tensor_dim1_stride lo
             0 };                  // tensor_dim1_stride hi
  v4i g2 = {0, 0, 0, 0};           // 2-D tensor: tile_dim3/4 = 0 (unused)
  v4i g3 = {0, 0, 0, 0};
#if __clang_major__ >= 23
  v8i gx = {0, 0, 0, 0, 0, 0, 0, 0};
  if (store) __builtin_amdgcn_tensor_store_from_lds(g0, g1, g2, g3, gx, 0);
  else       __builtin_amdgcn_tensor_load_to_lds(g0, g1, g2, g3, gx, 0);
#else
  if (store) __builtin_amdgcn_tensor_store_from_lds(g0, g1, g2, g3, 0);
  else       __builtin_amdgcn_tensor_load_to_lds(g0, g1, g2, g3, 0);
#endif
}
#endif

#define HW   512
#define NCH  48            // 16*3
#define NPIX 262144        // 512*512
#define C7   (1.0f/7.0f)
#define PI_F 3.14159265358979f

__device__ __forceinline__ int reflect512(int i) {
  if (i < 0) i = -i;
  if (i > 511) i = 1022 - i;
  return i;
}

__device__ __forceinline__ int quant16(float v) {
  int q = (int)(v * 15.0f);
  return q < 0 ? 0 : (q > 15 ? 15 : q);
}

// ---------------------------------------------------------------- zero counts
__global__ void zero_kernel(unsigned* p, int n) {
  int i = blockIdx.x * 256 + threadIdx.x;
  if (i < n) p[i] = 0u;
}

// ------------------------------------------------- histogram + GLCM (integer)
__global__ __launch_bounds__(256) void hist_kernel(const float* __restrict__ x,
                                                   unsigned* __restrict__ cntH,
                                                   unsigned* __restrict__ cntG) {
  int blk  = blockIdx.x;         // 48 chans * 16 bands
  int chan = blk >> 4;
  int band = blk & 15;
  const float* img = x + (size_t)chan * NPIX;
  __shared__ unsigned shH[256], shG[256];
  int t = threadIdx.x;
  shH[t] = 0u; shG[t] = 0u;
  __syncthreads();
  int y0 = band * 32;
  for (int i = t; i < 32 * 512; i += 256) {
    int y  = y0 + (i >> 9);
    int xx = i & 511;
    if (i + 2048 < 32 * 512)   // gfx1250 global_prefetch_b8
      __builtin_prefetch(img + (size_t)(y0 + ((i + 2048) >> 9)) * 512 + ((i + 2048) & 511), 0, 1);
    float v = img[(size_t)y * 512 + xx];
    int hb = (int)(v * 256.0f); hb = hb < 0 ? 0 : (hb > 255 ? 255 : hb);
    atomicAdd(&shH[hb], 1u);
    int q  = quant16(v);
    int ym = (y == 0) ? 511 : y - 1;
    int xm = (xx == 0) ? 511 : xx - 1;
    int xp = (xx == 511) ? 0 : xx + 1;
    int qW  = quant16(img[(size_t)y  * 512 + xm]);   // roll +1 on W  (0 deg)
    int qNE = quant16(img[(size_t)ym * 512 + xp]);   // 45 deg
    int qN  = quant16(img[(size_t)ym * 512 + xx]);   // 90 deg
    int qNW = quant16(img[(size_t)ym * 512 + xm]);   // 135 deg
    atomicAdd(&shG[q * 16 + qW],  1u);
    atomicAdd(&shG[q * 16 + qNE], 1u);
    atomicAdd(&shG[q * 16 + qN],  1u);
    atomicAdd(&shG[q * 16 + qNW], 1u);
  }
  __syncthreads();
  atomicAdd(&cntH[chan * 256 + t], shH[t]);
  atomicAdd(&cntG[chan * 256 + t], shG[t]);
}

// -------------------------------------------------------- per-channel entropy
__global__ __launch_bounds__(256) void entropy_kernel(const unsigned* __restrict__ cntH,
                                                      const unsigned* __restrict__ cntG,
                                                      float* __restrict__ ent,
                                                      float* __restrict__ glcm) {
  int chan = blockIdx.x, t = threadIdx.x;
  __shared__ float sh[256];
  float p = (float)cntH[chan * 256 + t] * (1.0f / (float)NPIX);
  sh[t] = -p * logf(p + 1e-10f);
  __syncthreads();
  for (int s = 128; s > 0; s >>= 1) { if (t < s) sh[t] += sh[t + s]; __syncthreads(); }
  if (t == 0) ent[chan] = fabsf(sh[0]);
  __syncthreads();
  float g = (float)cntG[chan * 256 + t] * (1.0f / (4.0f * (float)NPIX));
  sh[t] = -g * logf(g + 1e-10f);
  __syncthreads();
  for (int s = 128; s > 0; s >>= 1) { if (t < s) sh[t] += sh[t + s]; __syncthreads(); }
  if (t == 0) glcm[chan] = sh[0];
}

// ------------------------------------------------------ ent mean+std (scalar)
__global__ void entstats_kernel(const float* __restrict__ ent, float* __restrict__ scal) {
  __shared__ float s1[64], s2[64];
  int t = threadIdx.x;
  float v = (t < NCH) ? ent[t] : 0.0f;
  s1[t] = v; s2[t] = v * v;
  __syncthreads();
  for (int s = 32; s > 0; s >>= 1) {
    if (t < s) { s1[t] += s1[t + s]; s2[t] += s2[t + s]; }
    __syncthreads();
  }
  if (t == 0) {
    float m  = s1[0] / (float)NCH;
    float vv = s2[0] / (float)NCH - m * m;
    scal[0] = m + sqrtf(fmaxf(vv, 0.0f));     // ent_div = mean + std
  }
}

// -------------------- 7x7 box local variance via V_WMMA_F32_16X16X4_F32 tiles
__global__ __launch_bounds__(256) void var_kernel(const float* __restrict__ x,
                                                  float* __restrict__ varmap,
                                                  float* __restrict__ psum,
                                                  float* __restrict__ psumsq,
                                                  float* __restrict__ pmax) {
  __shared__ float lds[8 * 1344];         // per-wave: patch 24x24 | Hm 24x16 | H2 24x16
  __shared__ float rs[256], rq[256], rm[256];
  int tid  = threadIdx.x;
  int lane = tid & 31, wave = tid >> 5;
  int tile = blockIdx.x * 8 + wave;       // 49152 tiles total
  int chan = tile >> 10;
  int rem  = tile & 1023;
  int ty = rem >> 5, tx = rem & 31;
  const float* img = x + (size_t)chan * NPIX;
  float* patch = &lds[wave * 1344];
  float* Hm = patch + 576;
  float* H2 = Hm + 384;
  int y0 = ty * 16 - 3, x0 = tx * 16 - 3;
  for (int i = lane; i < 576; i += 32) {
    int pr = i / 24, pc = i - pr * 24;
    float v = 0.0f;
    if (pr < 22 && pc < 22)
      v = img[(size_t)reflect512(y0 + pr) * 512 + reflect512(x0 + pc)];
    patch[pr * 24 + pc] = v;
  }
  __syncthreads();
  int M  = lane & 15;
  int hi = lane >> 4;
  float lsum = 0.0f, lsq = 0.0f, lmx = -3.4e38f;
  float* vbase = varmap + (size_t)chan * NPIX;
#ifdef USE_WMMA_F32
  // Horizontal pass: H = P(24x24) @ Cmat(24x16), Cmat[k][n] = (n<=k<=n+6)?1/7:0
  v8f accTm = {}, accBm = {}, accTs = {}, accBs = {};
  for (int kb = 0; kb < 24; kb += 4) {
    int k0 = kb + hi * 2;
    v2f amT = { patch[M * 24 + k0],       patch[M * 24 + k0 + 1] };
    v2f amB = { patch[(M + 8) * 24 + k0], patch[(M + 8) * 24 + k0 + 1] };
    v2f asT = { amT.x * amT.x, amT.y * amT.y };
    v2f asB = { amB.x * amB.x, amB.y * amB.y };
    v2f b = { (k0     >= M && k0     <= M + 6) ? C7 : 0.0f,
              (k0 + 1 >= M && k0 + 1 <= M + 6) ? C7 : 0.0f };
    accTm = wmma4(amT, b, accTm);
    accBm = wmma4(amB, b, accBm);
    accTs = wmma4(asT, b, accTs);
    accBs = wmma4(asB, b, accBs);
  }
#pragma unroll
  for (int v = 0; v < 8; ++v) {           // rows 0..15 from top matmul
    int r = v + 8 * hi;
    Hm[r * 16 + M] = accTm[v];
    H2[r * 16 + M] = accTs[v];
  }
  if (hi) {                                // rows 16..23 from bottom matmul
#pragma unroll
    for (int v = 0; v < 8; ++v) {
      int r = v + 16;
      Hm[r * 16 + M] = accBm[v];
      H2[r * 16 + M] = accBs[v];
    }
  }
  __syncthreads();
  // Vertical pass: Out = Rmat(16x24) @ H(24x16), Rmat[m][k] = (m<=k<=m+6)?1/7:0
  v8f accVm = {}, accVs = {};
  for (int kb = 0; kb < 24; kb += 4) {
    int k0 = kb + hi * 2;
    v2f a = { (k0     >= M && k0     <= M + 6) ? C7 : 0.0f,
              (k0 + 1 >= M && k0 + 1 <= M + 6) ? C7 : 0.0f };
    v2f bm = { Hm[k0 * 16 + M], Hm[(k0 + 1) * 16 + M] };
    v2f bs = { H2[k0 * 16 + M], H2[(k0 + 1) * 16 + M] };
    accVm = wmma4(a, bm, accVm);
    accVs = wmma4(a, bs, accVs);
  }
#pragma unroll
  for (int v = 0; v < 8; ++v) {
    int r = v + 8 * hi;
    float m1 = accVm[v], m2 = accVs[v];
    float var = m2 - m1 * m1;
    vbase[(size_t)(ty * 16 + r) * 512 + (tx * 16 + M)] = var;
    lsum += var; lsq += var * var; lmx = fmaxf(lmx, var);
  }
#else
  for (int v = 0; v < 8; ++v) {
    int r = v + 8 * hi;
    float s1 = 0.0f, s2 = 0.0f;
    for (int i = 0; i < 7; ++i)
      for (int j = 0; j < 7; ++j) {
        float t2 = patch[(r + i) * 24 + (M + j)];
        s1 += t2; s2 += t2 * t2;
      }
    float m1 = s1 * (1.0f / 49.0f), m2 = s2 * (1.0f / 49.0f);
    float var = m2 - m1 * m1;
    vbase[(size_t)(ty * 16 + r) * 512 + (tx * 16 + M)] = var;
    lsum += var; lsq += var * var; lmx = fmaxf(lmx, var);
  }
#endif
  rs[tid] = lsum; rq[tid] = lsq; rm[tid] = lmx;
  __syncthreads();
  for (int s = 128; s > 0; s >>= 1) {
    if (tid < s) { rs[tid] += rs[tid + s]; rq[tid] += rq[tid + s]; rm[tid] = fmaxf(rm[tid], rm[tid + s]); }
    __syncthreads();
  }
  if (tid == 0) { psum[blockIdx.x] = rs[0]; psumsq[blockIdx.x] = rq[0]; pmax[blockIdx.x] = rm[0]; }
}

// --------------------------------------------------- var stats (deterministic)
__global__ __launch_bounds__(256) void varstats_kernel(const float* __restrict__ psum,
                                                       const float* __restrict__ psumsq,
                                                       const float* __restrict__ pmax,
                                                       float* __restrict__ scal) {
  __shared__ double ds1[256], ds2[256];
  __shared__ float dm[256];
  int t = threadIdx.x;
  double a = 0.0, b = 0.0; float mx = -3.4e38f;
  for (int i = t; i < 6144; i += 256) { a += psum[i]; b += psumsq[i]; mx = fmaxf(mx, pmax[i]); }
  ds1[t] = a; ds2[t] = b; dm[t] = mx;
  __syncthreads();
  for (int s = 128; s > 0; s >>= 1) {
    if (t < s) { ds1[t] += ds1[t + s]; ds2[t] += ds2[t + s]; dm[t] = fmaxf(dm[t], dm[t + s]); }
    __syncthreads();
  }
  if (t == 0) {
    double N    = (double)NCH * (double)NPIX;
    double mean = ds1[0] / N;
    double var  = ds2[0] / N - mean * mean; if (var < 0.0) var = 0.0;
    float vdiv  = (float)(mean - sqrt(var));     // var_div = mean - std
    scal[1] = vdiv;
    scal[2] = 1.0f / (dm[0] - vdiv);             // 1/(max - var_div)
  }
}

// ----------------------------- fused attention map: writes x*attn to staging
__global__ __launch_bounds__(256) void attn_kernel(const float* __restrict__ x,
                                                   const float* __restrict__ gab,
                                                   const float* __restrict__ varmap,
                                                   const float* __restrict__ ent,
                                                   const float* __restrict__ glcm,
                                                   const float* __restrict__ scal,
                                                   float* __restrict__ stage) {
  int chan = blockIdx.z;
  int tx0 = blockIdx.x * 16, ty0 = blockIdx.y * 16;
  int lx = threadIdx.x, ly = threadIdx.y;
  int tid = ly * 16 + lx;
  __shared__ float tile[20][20];
  __shared__ float gk[25];
  const float* img = x + (size_t)chan * NPIX;
  if (tid < 25) gk[tid] = gab[tid];
  for (int i = tid; i < 400; i += 256) {
    int r = i / 20, c = i - r * 20;
    tile[r][c] = img[(size_t)reflect512(ty0 + r - 2) * 512 + reflect512(tx0 + c - 2)];
  }
  __syncthreads();
  int gy = ty0 + ly, gx = tx0 + lx;
  float xc = tile[ly + 2][lx + 2];
  float n  = tile[ly + 1][lx + 2], s_ = tile[ly + 3][lx + 2];
  float w_ = tile[ly + 2][lx + 1], e  = tile[ly + 2][lx + 3];
  float nw = tile[ly + 1][lx + 1], ne = tile[ly + 1][lx + 3];
  float sw = tile[ly + 3][lx + 1], se = tile[ly + 3][lx + 3];
  float lap = fabsf(n + s_ + w_ + e - 4.0f * xc);                 // reflect pad
  float gxs = (ne + 2.0f * e + se) - (nw + 2.0f * w_ + sw);
  float gys = (sw + 2.0f * s_ + se) - (nw + 2.0f * n + ne);
  float sob = sqrtf(gxs * gxs + gys * gys + 1e-12f);
  float gb = 0.0f;                                                 // zero pad
#pragma unroll
  for (int i = 0; i < 5; ++i)
#pragma unroll
    for (int j = 0; j < 5; ++j) {
      int yy = gy + i - 2, xxg = gx + j - 2;
      float v = (yy >= 0 && yy < 512 && xxg >= 0 && xxg < 512) ? tile[ly + i][lx + j] : 0.0f;
      gb += gk[i * 5 + j] * v;
    }
  gb = fabsf(gb);
  size_t idx = (size_t)chan * NPIX + (size_t)gy * 512 + gx;
  float var  = varmap[idx];
  float entn = fminf(ent[chan] / scal[0], 1.0f);
  float varn = fmaxf((var - scal[1]) * scal[2], 1e-8f);
  float vs   = varn * sqrtf(fmaxf(1.0f - entn, 0.0f));            // beta = 0.5
  float attn = (vs + glcm[chan] + gb + lap + sob) * 0.2f;
  stage[idx] = xc * attn;
}

// --------------------------------------- Stockham radix-2 512-pt FFT (in LDS)
__device__ __forceinline__ void fft512(float2* bufA, float2* bufB, float sign, int t) {
  float2 *src = bufA, *dst = bufB;
#pragma unroll 1
  for (int stage = 0; stage < 9; ++stage) {
    int m  = 1 << stage;
    int p  = t >> stage;
    int mp = p << stage;
    float2 a = src[t];
    float2 b = src[t + 256];
    float2 apb = make_float2(a.x + b.x, a.y + b.y);
    float2 amb = make_float2(a.x - b.x, a.y - b.y);
    float ang = sign * PI_F * (float)p / (float)(256 >> stage);
    float sn, cs;
    __sincosf(ang, &sn, &cs);
    float2 tw = make_float2(cs * amb.x - sn * amb.y, cs * amb.y + sn * amb.x);
    dst[t + mp]     = apb;
    dst[t + mp + m] = tw;
    __syncthreads();
    float2* tmp = src; src = dst; dst = tmp;
  }
  // 9 stages: result lands in bufB
}

__global__ __launch_bounds__(256) void fft_row_fwd(const float* __restrict__ stage, float2* __restrict__ cbuf) {
  __shared__ float2 A[512], B[512];
  int row = blockIdx.x, t = threadIdx.x;
  const float* src = stage + (size_t)row * 512;
  A[t]       = make_float2(src[t], 0.0f);
  A[t + 256] = make_float2(src[t + 256], 0.0f);
  __syncthreads();
  fft512(A, B, -1.0f, t);
  float2* out = cbuf + (size_t)row * 512;
  out[t] = B[t]; out[t + 256] = B[t + 256];
}

__global__ __launch_bounds__(256) void fft_col(float2* __restrict__ cbuf) {
  __shared__ float2 A[512], B[512];
  int blk = blockIdx.x;
  int chan = blk >> 9, kx = blk & 511;
  float2* base = cbuf + (size_t)chan * NPIX + kx;
  int t = threadIdx.x;
#ifdef HAVE_TDM
  // Tensor Data Mover: gather strided column (1x512 tile, 8B elems, stride 512)
  if (t < 32) {  // wave 0 issues the DMA; TDM ignores EXEC, one op per block
    tdm_col((unsigned)(size_t)(void*)A, (unsigned long long)(size_t)base, false);
    __builtin_amdgcn_s_wait_tensorcnt(0);
  }
  __syncthreads();
#else
  A[t]       = base[(size_t)t * 512];
  A[t + 256] = base[(size_t)(t + 256) * 512];
  __syncthreads();
#endif
  fft512(A, B, -1.0f, t);                  // forward along H -> B
  float fx = (kx < 256) ? kx * (1.0f / 512.0f) : (kx - 512) * (1.0f / 512.0f);
  for (int ky = t; ky < 512; ky += 256) {
    float fy = (ky < 256) ? ky * (1.0f / 512.0f) : (ky - 512) * (1.0f / 512.0f);
    float k2 = fx * fx + fy * fy;
    if (kx == 0 && ky == 0) k2 = 1.0f;
    float sc = -1.0f / (4.0f * PI_F * PI_F * k2);
    B[ky].x *= sc; B[ky].y *= sc;
  }
  __syncthreads();
  fft512(B, A, +1.0f, t);                  // inverse along H -> A (barrier inside)
#ifdef HAVE_TDM
  if (t < 32) {  // scatter column back via TDM; ENDPGM implies wait-idle too
    tdm_col((unsigned)(size_t)(void*)A, (unsigned long long)(size_t)base, true);
    __builtin_amdgcn_s_wait_tensorcnt(0);
  }
#else
  base[(size_t)t * 512]         = A[t];
  base[(size_t)(t + 256) * 512] = A[t + 256];
#endif
}

__global__ __launch_bounds__(256) void fft_row_inv(const float2* __restrict__ cbuf, float* __restrict__ out) {
  __shared__ float2 A[512], B[512];
  int row = blockIdx.x, t = threadIdx.x;
  const float2* src = cbuf + (size_t)row * 512;
  A[t] = src[t]; A[t + 256] = src[t + 256];
  __syncthreads();
  fft512(A, B, +1.0f, t);
  const float inv = 1.0f / (512.0f * 512.0f);
  float* o = out + (size_t)row * 512;
  o[t]       = B[t].x * inv;
  o[t + 256] = B[t + 256].x * inv;
}

// -----------------------------------------------------------------------------
extern "C" void kernel_launch(void* const* d_in, const int* in_sizes, int n_in,
                              void* d_out, int out_size, void* d_ws, size_t ws_size,
                              hipStream_t stream) {
  (void)in_sizes; (void)n_in; (void)out_size; (void)ws_size;
  const float* x   = (const float*)d_in[0];
  const float* gab = (const float*)d_in[1];     // 5x5 gabor kernel
  char* ws = (char*)d_ws;
  // ws layout (bytes)
  unsigned* cntH   = (unsigned*)(ws + 0);            // 48*256 u32
  unsigned* cntG   = (unsigned*)(ws + 49152);        // 48*256 u32
  float*    ent    = (float*)(ws + 98304);           // 48
  float*    glcm   = (float*)(ws + 98560);           // 48
  float*    scal   = (float*)(ws + 98816);           // [0]=ent_div [1]=var_div [2]=1/denom
  float*    psum   = (float*)(ws + 99072);           // 6144
  float*    psumsq = psum + 6144;
  float*    pmax   = psumsq + 6144;
  float*    varmap = (float*)(ws + 172800);          // 48*512*512 f32
  float2*   cbuf   = (float2*)(ws + 50504448);       // 48*512*512 complex
  float*    stage  = (float*)d_out;                  // reuse d_out as real staging

  zero_kernel    <<<96,   256, 0, stream>>>(cntH, 24576);        // hist+glcm contiguous
  hist_kernel    <<<768,  256, 0, stream>>>(x, cntH, cntG);
  entropy_kernel <<<48,   256, 0, stream>>>(cntH, cntG, ent, glcm);
  entstats_kernel<<<1,     64, 0, stream>>>(ent, scal);
  var_kernel     <<<6144, 256, 0, stream>>>(x, varmap, psum, psumsq, pmax);
  varstats_kernel<<<1,    256, 0, stream>>>(psum, psumsq, pmax, scal);
  attn_kernel    <<<dim3(32, 32, 48), dim3(16, 16), 0, stream>>>(x, gab, varmap, ent, glcm, scal, stage);
  fft_row_fwd    <<<24576, 256, 0, stream>>>(stage, cbuf);
  fft_col        <<<24576, 256, 0, stream>>>(cbuf);
  fft_row_inv    <<<24576, 256, 0, stream>>>(cbuf, (float*)d_out);
}